// BilateralAugmentation_26121991094645
// MI455X (gfx1250) — compile-verified
//
#include <hip/hip_runtime.h>
#include <math.h>

#define Bb 4
#define Nn 8192
#define Kk 16
#define DIN 16
#define Hh 16
#define DOUT 32
#define EPS 1e-5f
#define BN_CNT  (Bb * Nn)        // 32768
#define BNK_CNT (Bb * Nn * Kk)   // 524288

typedef float v2f __attribute__((ext_vector_type(2)));
typedef float v8f __attribute__((ext_vector_type(8)));

// ---------------------------------------------------------------------------
// K0: zero the BN stats accumulators (5 stages x 32 floats: [16 sums][16 sumsq])
// ---------------------------------------------------------------------------
__global__ void zero_stats_kernel(float* __restrict__ stats) {
    int t = threadIdx.x;
    if (t < 160) stats[t] = 0.0f;
}

// ---------------------------------------------------------------------------
// K1a: q[i] = (-2x, -2y, -2z, |p|^2)  per point
// ---------------------------------------------------------------------------
__global__ __launch_bounds__(256) void qprep_kernel(const float* __restrict__ p,
                                                    float4* __restrict__ q) {
    int i = blockIdx.x * 256 + threadIdx.x;           // i < B*N
    float x = p[i * 3 + 0], y = p[i * 3 + 1], z = p[i * 3 + 2];
    q[i] = make_float4(-2.0f * x, -2.0f * y, -2.0f * z, x * x + y * y + z * z);
}

// ---------------------------------------------------------------------------
// K1b: fused KNN.  One wave owns 32 query rows (two 16-row A tiles).
// Per 16-col tile: two V_WMMA_F32_16X16X4_F32, C tiles -> per-wave LDS,
// each lane scans its own row and maintains a sorted top-16 in registers.
// grid = B * N / 256 = 128 blocks, block = 256 (8 waves)
// ---------------------------------------------------------------------------
__global__ __launch_bounds__(256) void knn_kernel(const float* __restrict__ p,
                                                  const float4* __restrict__ q,
                                                  int* __restrict__ idx) {
    __shared__ float cbuf[8 * 512];                   // 8 waves x (32 rows x 16 cols)

    const int wave = threadIdx.x >> 5;
    const int lane = threadIdx.x & 31;
    const int half = lane >> 4;                       // 0: K=0,1   1: K=2,3
    const int m    = lane & 15;

    const int b  = blockIdx.x >> 5;                   // 32 blocks per batch
    const int r0 = (blockIdx.x & 31) * 256 + wave * 32;

    const float*  pb = p + (size_t)b * Nn * 3;
    const float4* qb = q + (size_t)b * Nn;

    // A fragments (rows r0..r0+15 and r0+16..r0+31), row vector (x,y,z,1)
    v2f a0, a1;
    {
        int r = r0 + m;
        float x = pb[r * 3 + 0], y = pb[r * 3 + 1], z = pb[r * 3 + 2];
        a0[0] = half ? z : x;
        a0[1] = half ? 1.0f : y;
        r = r0 + 16 + m;
        x = pb[r * 3 + 0]; y = pb[r * 3 + 1]; z = pb[r * 3 + 2];
        a1[0] = half ? z : x;
        a1[1] = half ? 1.0f : y;
    }

    float bd[16];
    int   bi[16];
#pragma unroll
    for (int s = 0; s < 16; ++s) { bd[s] = 3.4e38f; bi[s] = 0; }

    float* cw = cbuf + wave * 512;

    for (int j0 = 0; j0 < Nn; j0 += 16) {
        // prefetch next column tile (L2-resident q array)
        int jp = j0 + 16 < Nn ? j0 + 16 : j0;
        __builtin_prefetch((const void*)(qb + jp + m), 0, 0);

        float4 qq = qb[j0 + m];
        v2f vb;
        vb[0] = half ? qq.z : qq.x;                   // K=0 / K=2
        vb[1] = half ? qq.w : qq.y;                   // K=1 / K=3

        v8f c0 = {}; v8f c1 = {};
        c0 = __builtin_amdgcn_wmma_f32_16x16x4_f32(false, a0, false, vb,
                                                   (short)0, c0, false, false);
        c1 = __builtin_amdgcn_wmma_f32_16x16x4_f32(false, a1, false, vb,
                                                   (short)0, c1, false, false);

        // spill C tiles: lane holds col n=m, rows (r + half*8)
#pragma unroll
        for (int r = 0; r < 8; ++r) {
            int mm = r + half * 8;
            cw[mm * 16 + m]        = c0[r];
            cw[(16 + mm) * 16 + m] = c1[r];
        }
        asm volatile("s_wait_dscnt 0" ::: "memory");

        // lane scans its own row (row index = lane within the wave's 32 rows)
#pragma unroll
        for (int n = 0; n < 16; ++n) {
            float d = cw[lane * 16 + n];
            int   j = j0 + n;
            if (d < bd[15]) {                          // strict < keeps tie stability
                bd[15] = d; bi[15] = j;
#pragma unroll
                for (int s = 15; s > 0; --s) {
                    if (bd[s] < bd[s - 1]) {
                        float td = bd[s]; bd[s] = bd[s - 1]; bd[s - 1] = td;
                        int   ti = bi[s]; bi[s] = bi[s - 1]; bi[s - 1] = ti;
                    }
                }
            }
        }
        asm volatile("s_wait_dscnt 0" ::: "memory");
    }

    const int r = r0 + lane;
#pragma unroll
    for (int s = 0; s < 16; ++s)
        idx[((size_t)b * Nn + r) * Kk + s] = bi[s];
}

// ---------------------------------------------------------------------------
// K2: y0 = f @ w0^T + b0  (per point), accumulate stage-0 BN stats
// grid = 128, block = 256 (thread per point)
// ---------------------------------------------------------------------------
__global__ __launch_bounds__(256) void k2_y0(const float* __restrict__ f,
                                             const float* __restrict__ w0,
                                             const float* __restrict__ b0,
                                             float* __restrict__ y0,
                                             float* __restrict__ stats) {
    __shared__ float w0s[256], b0s[16], ssum[32];
    int t = threadIdx.x;
    w0s[t] = w0[t];
    if (t < 16) b0s[t] = b0[t];
    if (t < 32) ssum[t] = 0.0f;
    __syncthreads();

    int r = blockIdx.x * 256 + t;
    float xr[16];
#pragma unroll
    for (int c = 0; c < 16; ++c) xr[c] = f[(size_t)r * 16 + c];
#pragma unroll
    for (int o = 0; o < 16; ++o) {
        float acc = b0s[o];
#pragma unroll
        for (int c = 0; c < 16; ++c) acc += xr[c] * w0s[o * 16 + c];
        y0[(size_t)r * 16 + o] = acc;
        atomicAdd(&ssum[o], acc);
        atomicAdd(&ssum[16 + o], acc * acc);
    }
    __syncthreads();
    if (t < 32) atomicAdd(&stats[t], ssum[t]);
}

// ---------------------------------------------------------------------------
// K3: f0 = relu(BN0(y0))     thread per element (524288)
// ---------------------------------------------------------------------------
__global__ __launch_bounds__(256) void k3_f0(const float* __restrict__ y0,
                                             const float* __restrict__ g0,
                                             const float* __restrict__ be0,
                                             const float* __restrict__ stats,
                                             float* __restrict__ f0) {
    int e = blockIdx.x * 256 + threadIdx.x;
    int o = e & 15;
    const float inv = 1.0f / (float)BN_CNT;
    float mu  = stats[o] * inv;
    float var = stats[16 + o] * inv - mu * mu;
    float sc  = g0[o] * rsqrtf(var + EPS);
    float v   = (y0[e] - mu) * sc + be0[o];
    f0[e] = fmaxf(v, 0.0f);
}

// ---------------------------------------------------------------------------
// K4: y1 = w1 @ lsc + b1  (lsc = [f0_c, f_knn - f0_c]), stage-1 stats
// thread per (b,n,k) element
// ---------------------------------------------------------------------------
__global__ __launch_bounds__(256) void k4_y1(const int* __restrict__ idx,
                                             const float* __restrict__ f0,
                                             const float* __restrict__ w1,
                                             const float* __restrict__ b1,
                                             float* __restrict__ y1,
                                             float* __restrict__ stats) {
    __shared__ float w1s[96], b1s[3], ssum[32];
    int t = threadIdx.x;
    if (t < 96) w1s[t] = w1[t];
    if (t < 3)  b1s[t] = b1[t];
    if (t < 32) ssum[t] = 0.0f;
    __syncthreads();

    int e  = blockIdx.x * 256 + t;
    int pt = e >> 4;
    int b  = pt >> 13;
    int j  = idx[e];
    const float* fc = f0 + (size_t)pt * 16;
    const float* fk = f0 + ((size_t)b * Nn + j) * 16;
    float lc[16], lk[16];
#pragma unroll
    for (int c = 0; c < 16; ++c) { lc[c] = fc[c]; lk[c] = fk[c] - lc[c]; }
#pragma unroll
    for (int o = 0; o < 3; ++o) {
        float acc = b1s[o];
#pragma unroll
        for (int c = 0; c < 16; ++c)
            acc += lc[c] * w1s[o * 32 + c] + lk[c] * w1s[o * 32 + 16 + c];
        y1[(size_t)e * 3 + o] = acc;
        atomicAdd(&ssum[o], acc);
        atomicAdd(&ssum[16 + o], acc * acc);
    }
    __syncthreads();
    if (t < 32) atomicAdd(&stats[32 + t], ssum[t]);
}

// ---------------------------------------------------------------------------
// K5: p_knn_tilde = relu(BN1(y1)) + p_knn  -> output; build lgc_aug(9);
//     on-the-fly y2/y3 for stage-2/3 stats (values recomputed later).
// ---------------------------------------------------------------------------
__global__ __launch_bounds__(256) void k5_pkt(const float* __restrict__ p,
                                              const int* __restrict__ idx,
                                              const float* __restrict__ y1,
                                              const float* __restrict__ w2,
                                              const float* __restrict__ b2,
                                              const float* __restrict__ w3,
                                              const float* __restrict__ b3,
                                              const float* __restrict__ g1,
                                              const float* __restrict__ be1,
                                              float* __restrict__ stats,
                                              float* __restrict__ out_pkt,
                                              float* __restrict__ lgc) {
    __shared__ float w2s[144], w3s[144], b2s[16], b3s[16], ssum[64];
    int t = threadIdx.x;
    if (t < 144) { w2s[t] = w2[t]; w3s[t] = w3[t]; }
    if (t < 16)  { b2s[t] = b2[t]; b3s[t] = b3[t]; }
    if (t < 64)  ssum[t] = 0.0f;
    __syncthreads();

    int e  = blockIdx.x * 256 + t;
    int pt = e >> 4;
    int b  = pt >> 13;
    int j  = idx[e];
    const float* s1 = stats + 32;
    const float inv = 1.0f / (float)BNK_CNT;
    const float* pc = p + (size_t)pt * 3;
    const float* pj = p + ((size_t)b * Nn + j) * 3;

    float la[9];
#pragma unroll
    for (int o = 0; o < 3; ++o) {
        float mu  = s1[o] * inv;
        float var = s1[16 + o] * inv - mu * mu;
        float sc  = g1[o] * rsqrtf(var + EPS);
        float v   = (y1[(size_t)e * 3 + o] - mu) * sc + be1[o];
        v = fmaxf(v, 0.0f) + pj[o];
        out_pkt[(size_t)e * 3 + o] = v;
        la[o]     = pc[o];
        la[o + 3] = pj[o] - pc[o];
        la[o + 6] = v;
    }
#pragma unroll
    for (int c = 0; c < 9; ++c) lgc[(size_t)e * 9 + c] = la[c];

#pragma unroll
    for (int o = 0; o < 16; ++o) {
        float a2 = b2s[o], a3 = b3s[o];
#pragma unroll
        for (int c = 0; c < 9; ++c) {
            a2 += la[c] * w2s[o * 9 + c];
            a3 += la[c] * w3s[o * 9 + c];
        }
        atomicAdd(&ssum[o], a2);      atomicAdd(&ssum[16 + o], a2 * a2);
        atomicAdd(&ssum[32 + o], a3); atomicAdd(&ssum[48 + o], a3 * a3);
    }
    __syncthreads();
    if (t < 64) atomicAdd(&stats[64 + t], ssum[t]);   // stages 2 (64..) and 3 (96..)
}

// ---------------------------------------------------------------------------
// K6: f_knn_tilde = relu(BN2(y2)) + f_knn  -> ws; y4 on-the-fly for stage-4 stats
// ---------------------------------------------------------------------------
__global__ __launch_bounds__(256) void k6_fkt(const int* __restrict__ idx,
                                              const float* __restrict__ f0,
                                              const float* __restrict__ lgc,
                                              const float* __restrict__ w2,
                                              const float* __restrict__ b2,
                                              const float* __restrict__ g2,
                                              const float* __restrict__ be2,
                                              const float* __restrict__ w4,
                                              const float* __restrict__ b4,
                                              float* __restrict__ stats,
                                              float* __restrict__ fkt) {
    __shared__ float w2s[144], b2s[16], w4s[768], b4s[16], ssum[32];
    int t = threadIdx.x;
    if (t < 144) w2s[t] = w2[t];
    if (t < 16)  { b2s[t] = b2[t]; b4s[t] = b4[t]; }
    for (int i = t; i < 768; i += 256) w4s[i] = w4[i];
    if (t < 32) ssum[t] = 0.0f;
    __syncthreads();

    int e  = blockIdx.x * 256 + t;
    int pt = e >> 4;
    int b  = pt >> 13;
    int j  = idx[e];
    const float* s2 = stats + 64;
    const float inv = 1.0f / (float)BNK_CNT;

    float la[9];
#pragma unroll
    for (int c = 0; c < 9; ++c) la[c] = lgc[(size_t)e * 9 + c];

    const float* fc = f0 + (size_t)pt * 16;
    const float* fkp = f0 + ((size_t)b * Nn + j) * 16;
    float lc[16], fk[16], fv[16];
#pragma unroll
    for (int c = 0; c < 16; ++c) { lc[c] = fc[c]; fk[c] = fkp[c]; }

#pragma unroll
    for (int o = 0; o < 16; ++o) {
        float a2 = b2s[o];
#pragma unroll
        for (int c = 0; c < 9; ++c) a2 += la[c] * w2s[o * 9 + c];
        float mu  = s2[o] * inv;
        float var = s2[16 + o] * inv - mu * mu;
        float sc  = g2[o] * rsqrtf(var + EPS);
        float v   = fmaxf((a2 - mu) * sc + be2[o], 0.0f) + fk[o];
        fv[o] = v;
        fkt[(size_t)e * 16 + o] = v;
    }
#pragma unroll
    for (int o = 0; o < 16; ++o) {
        float a4 = b4s[o];
#pragma unroll
        for (int c = 0; c < 16; ++c)
            a4 += lc[c] * w4s[o * 48 + c]
                + (fk[c] - lc[c]) * w4s[o * 48 + 16 + c]
                + fv[c] * w4s[o * 48 + 32 + c];
        atomicAdd(&ssum[o], a4);
        atomicAdd(&ssum[16 + o], a4 * a4);
    }
    __syncthreads();
    if (t < 32) atomicAdd(&stats[128 + t], ssum[t]);  // stage 4
}

// ---------------------------------------------------------------------------
// K7: alc = [relu(BN3(y3)), relu(BN4(y4))]  (y3 from lgc_aug, y4 from lsc_aug)
// ---------------------------------------------------------------------------
__global__ __launch_bounds__(256) void k7_alc(const int* __restrict__ idx,
                                              const float* __restrict__ f0,
                                              const float* __restrict__ lgc,
                                              const float* __restrict__ fkt,
                                              const float* __restrict__ w3,
                                              const float* __restrict__ b3,
                                              const float* __restrict__ g3,
                                              const float* __restrict__ be3,
                                              const float* __restrict__ w4,
                                              const float* __restrict__ b4,
                                              const float* __restrict__ g4,
                                              const float* __restrict__ be4,
                                              const float* __restrict__ stats,
                                              float* __restrict__ out_alc) {
    __shared__ float w3s[144], b3s[16], w4s[768], b4s[16];
    int t = threadIdx.x;
    if (t < 144) w3s[t] = w3[t];
    if (t < 16)  { b3s[t] = b3[t]; b4s[t] = b4[t]; }
    for (int i = t; i < 768; i += 256) w4s[i] = w4[i];
    __syncthreads();

    int e  = blockIdx.x * 256 + t;
    int pt = e >> 4;
    int b  = pt >> 13;
    int j  = idx[e];
    const float* s3 = stats + 96;
    const float* s4 = stats + 128;
    const float inv = 1.0f / (float)BNK_CNT;

    float la[9];
#pragma unroll
    for (int c = 0; c < 9; ++c) la[c] = lgc[(size_t)e * 9 + c];
    const float* fc = f0 + (size_t)pt * 16;
    const float* fkp = f0 + ((size_t)b * Nn + j) * 16;
    float lc[16], fk[16], fv[16];
#pragma unroll
    for (int c = 0; c < 16; ++c) {
        lc[c] = fc[c]; fk[c] = fkp[c];
        fv[c] = fkt[(size_t)e * 16 + c];
    }

#pragma unroll
    for (int o = 0; o < 16; ++o) {
        float a3 = b3s[o];
#pragma unroll
        for (int c = 0; c < 9; ++c) a3 += la[c] * w3s[o * 9 + c];
        float mu  = s3[o] * inv;
        float var = s3[16 + o] * inv - mu * mu;
        float sc  = g3[o] * rsqrtf(var + EPS);
        out_alc[(size_t)e * 32 + o] = fmaxf((a3 - mu) * sc + be3[o], 0.0f);

        float a4 = b4s[o];
#pragma unroll
        for (int c = 0; c < 16; ++c)
            a4 += lc[c] * w4s[o * 48 + c]
                + (fk[c] - lc[c]) * w4s[o * 48 + 16 + c]
                + fv[c] * w4s[o * 48 + 32 + c];
        mu  = s4[o] * inv;
        var = s4[16 + o] * inv - mu * mu;
        sc  = g4[o] * rsqrtf(var + EPS);
        out_alc[(size_t)e * 32 + 16 + o] = fmaxf((a4 - mu) * sc + be4[o], 0.0f);
    }
}

// ---------------------------------------------------------------------------
extern "C" void kernel_launch(void* const* d_in, const int* in_sizes, int n_in,
                              void* d_out, int out_size, void* d_ws, size_t ws_size,
                              hipStream_t stream) {
    const float* p  = (const float*)d_in[0];
    const float* f  = (const float*)d_in[1];
    const float* w0 = (const float*)d_in[2];
    const float* b0 = (const float*)d_in[3];
    const float* g0 = (const float*)d_in[4];
    const float* be0 = (const float*)d_in[5];
    const float* w1 = (const float*)d_in[6];
    const float* b1 = (const float*)d_in[7];
    const float* g1 = (const float*)d_in[8];
    const float* be1 = (const float*)d_in[9];
    const float* w2 = (const float*)d_in[10];
    const float* b2 = (const float*)d_in[11];
    const float* g2 = (const float*)d_in[12];
    const float* be2 = (const float*)d_in[13];
    const float* w3 = (const float*)d_in[14];
    const float* b3 = (const float*)d_in[15];
    const float* g3 = (const float*)d_in[16];
    const float* be3 = (const float*)d_in[17];
    const float* w4 = (const float*)d_in[18];
    const float* b4 = (const float*)d_in[19];
    const float* g4 = (const float*)d_in[20];
    const float* be4 = (const float*)d_in[21];

    float* out_alc = (float*)d_out;                              // B*N*K*32
    float* out_pkt = out_alc + (size_t)BNK_CNT * 32;             // B*N*K*3

    // workspace layout
    char* w = (char*)d_ws;
    int*    idx = (int*)w;        w += (size_t)BNK_CNT * sizeof(int);      // 2 MB
    float4* q   = (float4*)w;     w += (size_t)BN_CNT * sizeof(float4);    // 0.5 MB
    float*  y0  = (float*)w;      w += (size_t)BN_CNT * 16 * sizeof(float);
    float*  f0  = (float*)w;      w += (size_t)BN_CNT * 16 * sizeof(float);
    float*  y1  = (float*)w;      w += (size_t)BNK_CNT * 3 * sizeof(float);
    float*  lgc = (float*)w;      w += (size_t)BNK_CNT * 9 * sizeof(float);
    float*  fkt = (float*)w;      w += (size_t)BNK_CNT * 16 * sizeof(float);
    float*  stats = (float*)w;    // 160 floats

    zero_stats_kernel<<<1, 256, 0, stream>>>(stats);
    qprep_kernel<<<BN_CNT / 256, 256, 0, stream>>>(p, q);
    knn_kernel<<<(Bb * Nn) / 256, 256, 0, stream>>>(p, q, idx);
    k2_y0<<<BN_CNT / 256, 256, 0, stream>>>(f, w0, b0, y0, stats);
    k3_f0<<<(BN_CNT * 16) / 256, 256, 0, stream>>>(y0, g0, be0, stats, f0);
    k4_y1<<<BNK_CNT / 256, 256, 0, stream>>>(idx, f0, w1, b1, y1, stats);
    k5_pkt<<<BNK_CNT / 256, 256, 0, stream>>>(p, idx, y1, w2, b2, w3, b3,
                                              g1, be1, stats, out_pkt, lgc);
    k6_fkt<<<BNK_CNT / 256, 256, 0, stream>>>(idx, f0, lgc, w2, b2, g2, be2,
                                              w4, b4, stats, fkt);
    k7_alc<<<BNK_CNT / 256, 256, 0, stream>>>(idx, f0, lgc, fkt, w3, b3, g3, be3,
                                              w4, b4, g4, be4, stats, out_alc);
}